// TopKLayer_23940147708126
// MI455X (gfx1250) — compile-verified
//
#include <hip/hip_runtime.h>
#include <stdint.h>

// ---------------------------------------------------------------------------
// MI455X top-k (k=64) over 4096 rows of 32768 fp32.
// Roofline: 512 MB input / 23.3 TB/s => ~22 us floor. Read HBM exactly once via
// gfx1250 async global->LDS copies (ASYNCcnt, TH_LOAD_NT), then exact radix
// select out of LDS. Only TWO full LDS scans per row (hist + candidate
// compaction); the remaining radix passes and the collect run over a ~800-entry
// compacted buffer, so aggregate LDS traffic stays under the HBM floor.
// ~146 KB LDS/block, 512 threads -> 2 blocks/WGP (<= 320 KB, 8 waves/SIMD) so
// one block's async staging overlaps the other block's LDS compute.
// NOTE: dynamic-LDS base is force-aligned to 16 B (async b128 stores require
// it); launch passes 16 B of slack.
// ---------------------------------------------------------------------------

#define TOPK 64
#define CAP  2048                   // candidate buffer capacity (exact fallback if exceeded)
static constexpr int kN = 32768;    // row length (fixed by reference)
static constexpr int kB = 512;      // threads per block = 16 wave32 waves

typedef __attribute__((address_space(3))) char lds_char_t;

__device__ __forceinline__ uint32_t lds_byte_off(void* p) {
  return (uint32_t)(uintptr_t)(lds_char_t*)p;
}

// Order-preserving float<->uint keys: larger float => larger key.
__device__ __forceinline__ uint32_t f2key(float f) {
  uint32_t u = __float_as_uint(f);
  return (u & 0x80000000u) ? ~u : (u | 0x80000000u);
}
__device__ __forceinline__ float key2f(uint32_t k) {
  uint32_t u = (k & 0x80000000u) ? (k ^ 0x80000000u) : ~k;
  return __uint_as_float(u);
}

// Wave32 ballot-aggregated LDS histogram add: lanes sharing a digit elect a
// leader that adds the peer count once (randn keys hit only a few exponent
// bins, so naive per-lane atomics would serialize).
__device__ __forceinline__ void hist_add_wave_agg(uint32_t* hist, uint32_t digit) {
  uint32_t active = (uint32_t)__ballot(1);
  uint32_t peers = active;
#pragma unroll
  for (int b = 0; b < 8; ++b) {
    uint32_t vote = (uint32_t)__ballot((int)((digit >> b) & 1u));
    peers &= ((digit >> b) & 1u) ? vote : ~vote;
  }
  peers &= active;
  uint32_t lane = (uint32_t)(threadIdx.x & 31u);
  if (lane == (uint32_t)(__ffs(peers) - 1))
    atomicAdd(&hist[digit], (uint32_t)__popc(peers));
}

__global__ __launch_bounds__(kB) void topk_row_kernel(const float* __restrict__ x,
                                                      float* __restrict__ out,
                                                      float* __restrict__ rowsum) {
  extern __shared__ __attribute__((aligned(16))) float dynLds[];  // kN floats + 16B slack
  __shared__ uint32_t hist[256];
  __shared__ uint32_t candKey[CAP];        // 8 KB
  __shared__ uint32_t candIdx[CAP];        // 8 KB
  __shared__ float    selVal[TOPK];
  __shared__ uint32_t selIdx[TOPK];
  __shared__ float    sortedVal[TOPK];
  __shared__ uint32_t s_prefix, s_need, s_cntG, s_cntE, s_candCnt;

  const int tid = threadIdx.x;
  const int row = blockIdx.x;
  const float* rowp = x + (size_t)row * kN;   // uniform -> SGPR pair

  // ---- Force 16B alignment of the staging buffer (async b128 stores) ------
  uint32_t base0   = lds_byte_off((void*)dynLds);
  uint32_t ldsBase = (base0 + 15u) & ~15u;                 // aligned byte offset
  float*   vals    = (float*)((char*)dynLds + (ldsBase - base0));

  // ---- Stage row into LDS: gfx1250 async copies, non-temporal (streamed) ---
  for (int c = tid; c < kN / 4; c += kB) {    // 8192 x 16B chunks
    uint32_t lOff = ldsBase + (uint32_t)c * 16u;
    uint32_t gOff = (uint32_t)c * 16u;
    asm volatile("global_load_async_to_lds_b128 %0, %1, %2 th:TH_LOAD_NT"
                 :
                 : "v"(lOff), "v"(gOff), "s"(rowp)
                 : "memory");
  }
  if (tid == 0) { s_prefix = 0u; s_cntG = 0u; s_cntE = 0u; s_candCnt = 0u; }
  asm volatile("s_wait_asynccnt 0" ::: "memory");  // my waves' copies done
  __syncthreads();                                  // everyone's copies done

  // ---- Full scan 1: histogram of top byte ---------------------------------
  if (tid < 256) hist[tid] = 0u;
  __syncthreads();
  for (int i = tid; i < kN; i += kB)
    hist_add_wave_agg(hist, f2key(vals[i]) >> 24);
  __syncthreads();
  if (tid == 0) {
    uint32_t cum = 0u, dSel = 0u;
    for (int d = 255; d >= 0; --d) {
      cum += hist[d];
      if (cum >= TOPK) { dSel = (uint32_t)d; break; }
    }
    s_need   = TOPK - (cum - hist[dSel]);
    s_prefix = dSel << 24;
  }
  __syncthreads();

  // ---- Full scan 2: compact candidates (top byte >= selected bin) ---------
  // Winners all satisfy key >= T and T's top byte == bTop, so winners are a
  // subset of the compacted set. Expected |cand| ~ 800 for randn.
  const uint32_t bTop = s_prefix >> 24;
  for (int i = tid; i < kN; i += kB) {
    uint32_t key = f2key(vals[i]);
    if ((key >> 24) >= bTop) {
      uint32_t p = atomicAdd(&s_candCnt, 1u);
      if (p < (uint32_t)CAP) { candKey[p] = key; candIdx[p] = (uint32_t)i; }
    }
  }
  __syncthreads();
  const uint32_t cnt  = s_candCnt;
  const bool small    = (cnt <= (uint32_t)CAP);   // exact fallback otherwise

  // ---- Radix passes 2..4 over candidates (or full fallback) ---------------
  for (int shift = 16; shift >= 0; shift -= 8) {
    if (tid < 256) hist[tid] = 0u;
    __syncthreads();
    uint32_t pref  = s_prefix;
    uint32_t hmask = 0xFFFFFFFFu << (shift + 8);  // bytes already decided
    if (small) {
      for (uint32_t i = tid; i < cnt; i += kB) {
        uint32_t key = candKey[i];
        if ((key & hmask) == pref) atomicAdd(&hist[(key >> shift) & 255u], 1u);
      }
    } else {
      for (int i = tid; i < kN; i += kB) {
        uint32_t key = f2key(vals[i]);
        if ((key & hmask) == pref) atomicAdd(&hist[(key >> shift) & 255u], 1u);
      }
    }
    __syncthreads();
    if (tid == 0) {
      uint32_t need = s_need;
      uint32_t cum = 0u, dSel = 0u;
      for (int d = 255; d >= 0; --d) {
        cum += hist[d];
        if (cum >= need) { dSel = (uint32_t)d; break; }
      }
      s_need   = need - (cum - hist[dSel]);
      s_prefix = pref | (dSel << shift);
    }
    __syncthreads();
  }

  const uint32_t T    = s_prefix;        // exact key of the k-th largest
  const uint32_t need = s_need;          // # of key==T elements to accept
  const uint32_t numG = TOPK - need;     // # of key> T elements (exact)

  // ---- Collect the 64 winners ---------------------------------------------
  if (small) {
    for (uint32_t i = tid; i < cnt; i += kB) {
      uint32_t key = candKey[i];
      if (key > T) {
        uint32_t p = atomicAdd(&s_cntG, 1u);
        selVal[p] = key2f(key); selIdx[p] = candIdx[i];
      } else if (key == T) {
        uint32_t p = atomicAdd(&s_cntE, 1u);
        if (p < need) { selVal[numG + p] = key2f(key); selIdx[numG + p] = candIdx[i]; }
      }
    }
  } else {
    for (int i = tid; i < kN; i += kB) {
      float v = vals[i];
      uint32_t key = f2key(v);
      if (key > T) {
        uint32_t p = atomicAdd(&s_cntG, 1u);
        selVal[p] = v; selIdx[p] = (uint32_t)i;
      } else if (key == T) {
        uint32_t p = atomicAdd(&s_cntE, 1u);
        if (p < need) { selVal[numG + p] = v; selIdx[numG + p] = (uint32_t)i; }
      }
    }
  }
  __syncthreads();

  // ---- O(64^2) rank sort (desc value, asc index) and emit -----------------
  const size_t obase = (size_t)row * (2 * TOPK);
  if (tid < TOPK) {
    float v = selVal[tid]; uint32_t id = selIdx[tid];
    int rank = 0;
#pragma unroll 8
    for (int j = 0; j < TOPK; ++j) {
      float vj = selVal[j]; uint32_t ij = selIdx[j];
      rank += (vj > v || (vj == v && ij < id)) ? 1 : 0;
    }
    sortedVal[rank] = v;
    out[obase + rank]        = (float)id;  // indices half
    out[obase + TOPK + rank] = v;          // raw values; normalized in pass 3
  }
  __syncthreads();
  if (tid == 0) {                          // deterministic rank-order sum
    float s = 0.f;
#pragma unroll 8
    for (int j = 0; j < TOPK; ++j) s += fabsf(sortedVal[j]);
    rowsum[row] = s;
  }
}

// Deterministic single-block tree reduction of per-row |v| sums -> global l1.
__global__ __launch_bounds__(1024) void l1_reduce_kernel(const float* __restrict__ rowsum,
                                                         float* __restrict__ l1,
                                                         int rows) {
  __shared__ float sm[1024];
  float s = 0.f;
  for (int i = threadIdx.x; i < rows; i += 1024) s += rowsum[i];
  sm[threadIdx.x] = s;
  __syncthreads();
  for (int off = 512; off > 0; off >>= 1) {
    if (threadIdx.x < off) sm[threadIdx.x] += sm[threadIdx.x + off];
    __syncthreads();
  }
  if (threadIdx.x == 0) *l1 = sm[0];
}

// Scale the value half of the output by 1/(l1 + 1e-6). Kernel 1 rewrites raw
// values each launch, so this in-place update is graph-replay safe.
__global__ void normalize_kernel(float* __restrict__ out,
                                 const float* __restrict__ l1,
                                 int total) {
  int i = blockIdx.x * blockDim.x + threadIdx.x;
  if (i >= total) return;
  float inv = 1.0f / (*l1 + 1e-6f);
  int row = i >> 6, j = i & 63;
  out[(size_t)row * (2 * TOPK) + TOPK + j] *= inv;
}

extern "C" void kernel_launch(void* const* d_in, const int* in_sizes, int n_in,
                              void* d_out, int out_size, void* d_ws, size_t ws_size,
                              hipStream_t stream) {
  (void)n_in; (void)out_size; (void)ws_size;
  const float* x = (const float*)d_in[0];   // [rows, 32768] fp32
  // d_in[1] is k (==64), fixed by the reference; TOPK is compile-time.
  float* out = (float*)d_out;               // [rows, 128] fp32
  int rows = in_sizes[0] / kN;              // 4096

  float* rowsum = (float*)d_ws;             // rows floats
  float* l1     = rowsum + rows;            // 1 float

  // +16 B dynamic-LDS slack for in-kernel 16B re-alignment of the staging buf.
  topk_row_kernel<<<rows, kB, kN * sizeof(float) + 16, stream>>>(x, out, rowsum);
  l1_reduce_kernel<<<1, 1024, 0, stream>>>(rowsum, l1, rows);

  int total = rows * TOPK;
  normalize_kernel<<<(total + 255) / 256, 256, 0, stream>>>(out, l1, total);
}